// LDM_91199335563503
// MI455X (gfx1250) — compile-verified
//
#include <hip/hip_runtime.h>

#define B_SEQ 128
#define T_STEPS 1000
#define DA 32
#define DX 16

typedef float v2f __attribute__((ext_vector_type(2)));
typedef float v8f __attribute__((ext_vector_type(8)));

// ---- WMMA f32 16x16x4 fragment helpers (wave32) ----------------------------
// A (16x4, MxK): lane 0-15 -> M, VGPR r + lane-half -> K = k0 + 2*hi + r
__device__ __forceinline__ v2f ldA(const float* M, int ld, int k0, int lane,
                                   int trans, float scale) {
  int m  = lane & 15;
  int kk = k0 + ((lane >> 4) << 1);
  v2f a;
  if (!trans) { a.x = M[m * ld + kk] * scale;  a.y = M[m * ld + kk + 1] * scale; }
  else        { a.x = M[kk * ld + m] * scale;  a.y = M[(kk + 1) * ld + m] * scale; }
  return a;
}
// B (4x16, KxN): lane 0-15 -> N, VGPR r + lane-half -> K (mirror of A)
__device__ __forceinline__ v2f ldB(const float* M, int ld, int k0, int lane,
                                   int trans) {
  int n  = lane & 15;
  int kk = k0 + ((lane >> 4) << 1);
  v2f b;
  if (!trans) { b.x = M[kk * ld + n];  b.y = M[(kk + 1) * ld + n]; }
  else        { b.x = M[n * ld + kk];  b.y = M[n * ld + kk + 1]; }
  return b;
}
// C/D (16x16 f32): VGPR r, lanes 0-15 -> (M=r, N=lane); lanes 16-31 -> (M=r+8)
__device__ __forceinline__ v8f ldC16(const float* M, int ld, int lane) {
  int n = lane & 15, mh = (lane >> 4) << 3;
  v8f c;
#pragma unroll
  for (int r = 0; r < 8; ++r) c[r] = M[(mh + r) * ld + n];
  return c;
}
__device__ __forceinline__ void stD16(float* M, int ld, v8f d, int lane) {
  int n = lane & 15, mh = (lane >> 4) << 3;
#pragma unroll
  for (int r = 0; r < 8; ++r) M[(mh + r) * ld + n] = d[r];
}

__device__ __forceinline__ v8f wmma4(v2f a, v2f b, v8f c) {
  return __builtin_amdgcn_wmma_f32_16x16x4_f32(false, a, false, b, (short)0, c,
                                               false, false);
}

// Register-resident 16x16 operand: 4 K-slices of v2f fragments
struct Frag4 { v2f s0, s1, s2, s3; };

__device__ __forceinline__ Frag4 loadAfrag16(const float* M, int ld, int trans,
                                             float scale, int lane, int kbase = 0) {
  Frag4 f;
  f.s0 = ldA(M, ld, kbase + 0,  lane, trans, scale);
  f.s1 = ldA(M, ld, kbase + 4,  lane, trans, scale);
  f.s2 = ldA(M, ld, kbase + 8,  lane, trans, scale);
  f.s3 = ldA(M, ld, kbase + 12, lane, trans, scale);
  return f;
}
__device__ __forceinline__ Frag4 loadBfrag16(const float* M, int ld, int trans,
                                             int lane, int kbase = 0) {
  Frag4 f;
  f.s0 = ldB(M, ld, kbase + 0,  lane, trans);
  f.s1 = ldB(M, ld, kbase + 4,  lane, trans);
  f.s2 = ldB(M, ld, kbase + 8,  lane, trans);
  f.s3 = ldB(M, ld, kbase + 12, lane, trans);
  return f;
}
// acc += A(16x16) * B(16x16): 4 chained v_wmma_f32_16x16x4_f32
__device__ __forceinline__ v8f gemmFF(const Frag4& a, const Frag4& b, v8f acc) {
  acc = wmma4(a.s0, b.s0, acc);
  acc = wmma4(a.s1, b.s1, acc);
  acc = wmma4(a.s2, b.s2, acc);
  acc = wmma4(a.s3, b.s3, acc);
  return acc;
}

// ---- Forward Kalman filter: one wave32 per sequence ------------------------
__global__ __launch_bounds__(32)
void ldm_filter(const float* __restrict__ g_a,   const float* __restrict__ g_mask,
                const float* __restrict__ g_A,   const float* __restrict__ g_C,
                const float* __restrict__ g_mu0, const float* __restrict__ g_L0,
                const float* __restrict__ g_Wlog,const float* __restrict__ g_Rlog,
                float* __restrict__ o_mupred, float* __restrict__ o_mut,
                float* __restrict__ o_Ppred,  float* __restrict__ o_Pt) {
  __shared__ float sA[DX * DX];
  __shared__ float sC[DA * DX];
  __shared__ float sP[DX * DX];     // P_pred carry
  __shared__ float sPt[DX * DX];
  __shared__ float sCP[DA * DX];
  __shared__ float sTmp[DX * DX];
  __shared__ float sAug[DA * 49];   // [S | CP] augmented, pad 49 (conflict-free)
  __shared__ float sW[DX];
  __shared__ float sR[DA];
  __shared__ float sMu[DX];         // mu_pred carry
  __shared__ float sMuT[DX];
  __shared__ float sInnov[DA];

  const int b = blockIdx.x;
  const int lane = threadIdx.x;

  for (int i = lane; i < DX * DX; i += 32) { sA[i] = g_A[i]; sP[i] = g_L0[i]; }
  for (int i = lane; i < DA * DX; i += 32) sC[i] = g_C[i];
  if (lane < DX) { sW[lane] = __expf(g_Wlog[lane]); sMu[lane] = g_mu0[lane]; }
  sR[lane] = __expf(g_Rlog[lane]);
  __syncthreads();

  // Loop-invariant operand fragments, resident in VGPRs for all 1000 steps.
  const Frag4 aC0  = loadAfrag16(sC,           DX, 0, 1.f, lane); // C rows 0-15
  const Frag4 aC1  = loadAfrag16(sC + 16 * DX, DX, 0, 1.f, lane); // C rows 16-31
  const Frag4 bCT0 = loadBfrag16(sC,           DX, 1, lane);      // C^T cols 0-15
  const Frag4 bCT1 = loadBfrag16(sC + 16 * DX, DX, 1, lane);      // C^T cols 16-31
  const Frag4 aA   = loadAfrag16(sA,           DX, 0, 1.f, lane); // A
  const Frag4 bAT  = loadBfrag16(sA,           DX, 1, lane);      // A^T
  v8f wdiag;                                                       // diag(W) frag
  {
    int n = lane & 15, mh = (lane >> 4) << 3;
#pragma unroll
    for (int r = 0; r < 8; ++r) wdiag[r] = (mh + r == n) ? sW[n] : 0.f;
  }

  const v8f vzero = {};
#pragma unroll 1
  for (int t = 0; t < T_STEPS; ++t) {
    const float* at = g_a + ((size_t)b * T_STEPS + t) * DA;
    const float  mt = g_mask[(size_t)b * T_STEPS + t];
    if (t + 1 < T_STEPS) __builtin_prefetch(at + DA, 0, 3);

    // CP = C * P_pred (32x16): shared B-fragment of P for both row blocks
    const Frag4 bP = loadBfrag16(sP, DX, 0, lane);
    v8f cp0 = gemmFF(aC0, bP, vzero);
    v8f cp1 = gemmFF(aC1, bP, vzero);
    stD16(sCP,           DX, cp0, lane);
    stD16(sCP + 16 * DX, DX, cp1, lane);
    __syncthreads();

    // S = CP * C^T (32x32) -> aug cols 0..31
    const Frag4 aCP0 = loadAfrag16(sCP,           DX, 0, 1.f, lane);
    const Frag4 aCP1 = loadAfrag16(sCP + 16 * DX, DX, 0, 1.f, lane);
    stD16(&sAug[0],            49, gemmFF(aCP0, bCT0, vzero), lane);
    stD16(&sAug[16],           49, gemmFF(aCP0, bCT1, vzero), lane);
    stD16(&sAug[16 * 49],      49, gemmFF(aCP1, bCT0, vzero), lane);
    stD16(&sAug[16 * 49 + 16], 49, gemmFF(aCP1, bCT1, vzero), lane);
    __syncthreads();

    // S += R (diag); RHS cols 32..47 = CP; innov = a_t - C mu_pred
    sAug[lane * 49 + lane] += sR[lane];
#pragma unroll
    for (int j = 0; j < DX; ++j) sAug[lane * 49 + 32 + j] = sCP[lane * DX + j];
    {
      float acc = at[lane];
#pragma unroll
      for (int j = 0; j < DX; ++j) acc -= sC[lane * DX + j] * sMu[j];
      sInnov[lane] = acc;
    }
    __syncthreads();

    // Gauss-Jordan (lane = row): X = S^{-1} CP  (32x16)
#pragma unroll 1
    for (int p = 0; p < DA; ++p) {
      float f = sAug[lane * 49 + p] / sAug[p * 49 + p];
      if (lane != p) {
#pragma unroll 4
        for (int c = p + 1; c < 48; ++c)
          sAug[lane * 49 + c] -= f * sAug[p * 49 + c];
      }
      __syncthreads();
    }
    {
      float inv = 1.0f / sAug[lane * 49 + lane];
#pragma unroll
      for (int j = 0; j < DX; ++j) sAug[lane * 49 + 32 + j] *= inv;
    }
    __syncthreads();
    // X at &sAug[32], ld 49:  K = mt * X^T (16x32)

    // mu_t = mu_pred + mt * X^T innov
    if (lane < DX) {
      float acc = 0.f;
#pragma unroll
      for (int j = 0; j < DA; ++j) acc += sAug[j * 49 + 32 + lane] * sInnov[j];
      sMuT[lane] = sMu[lane] + mt * acc;
    }

    // P_t = P_pred - K*CP   (K = mt*X^T, 16x32; mask folded into A-frag scale)
    const Frag4 aK0  = loadAfrag16(&sAug[32], 49, 1, mt, lane, 0);
    const Frag4 aK1  = loadAfrag16(&sAug[32], 49, 1, mt, lane, 16);
    const Frag4 bCP0 = loadBfrag16(sCP, DX, 0, lane, 0);
    const Frag4 bCP1 = loadBfrag16(sCP, DX, 0, lane, 16);
    v8f kcp = gemmFF(aK1, bCP1, gemmFF(aK0, bCP0, vzero));
    v8f pt  = ldC16(sP, DX, lane) - kcp;
    stD16(sPt, DX, pt, lane);
    __syncthreads();

    // mu_next = A * mu_t
    if (lane < DX) {
      float acc = 0.f;
#pragma unroll
      for (int j = 0; j < DX; ++j) acc += sA[lane * DX + j] * sMuT[j];
      sMu[lane] = acc;
    }

    // P_next = A * P_t * A^T + W  (W-diag as the initial C-fragment)
    const Frag4 bPt = loadBfrag16(sPt, DX, 0, lane);
    v8f t1 = gemmFF(aA, bPt, vzero);
    stD16(sTmp, DX, t1, lane);
    __syncthreads();
    const Frag4 aT1 = loadAfrag16(sTmp, DX, 0, 1.f, lane);
    v8f pn = gemmFF(aT1, bAT, wdiag);
    __syncthreads();
    stD16(sP, DX, pn, lane);
    __syncthreads();

    // emit step outputs: mu_pred_all[t]=mu_next, P_pred_all[t]=P_next
    size_t mu_off = ((size_t)b * T_STEPS + t) * DX;
    size_t P_off  = ((size_t)b * T_STEPS + t) * DX * DX;
    if (lane < DX) {
      o_mupred[mu_off + lane] = sMu[lane];
      o_mut[mu_off + lane]    = sMuT[lane];
    }
#pragma unroll
    for (int r = 0; r < 8; ++r) {
      o_Ppred[P_off + lane * 8 + r] = sP[lane * 8 + r];
      o_Pt[P_off + lane * 8 + r]    = sPt[lane * 8 + r];
    }
    __syncthreads();
  }
}

// ---- RTS backward smoother: one wave32 per sequence ------------------------
__global__ __launch_bounds__(32)
void ldm_smooth(const float* __restrict__ g_A, const int* __restrict__ g_flag,
                const float* __restrict__ o_mupred, const float* __restrict__ o_mut,
                const float* __restrict__ o_Ppred,  const float* __restrict__ o_Pt,
                float* __restrict__ o_muback, float* __restrict__ o_Pback) {
  const int b = blockIdx.x;
  const int lane = threadIdx.x;

  if (*g_flag == 0) {  // reference: ones when smoothing disabled
#pragma unroll 1
    for (int t = 0; t < T_STEPS; ++t) {
      size_t mu_off = ((size_t)b * T_STEPS + t) * DX;
      size_t P_off  = ((size_t)b * T_STEPS + t) * DX * DX;
      if (lane < DX) o_muback[mu_off + lane] = 1.0f;
#pragma unroll
      for (int r = 0; r < 8; ++r) o_Pback[P_off + lane * 8 + r] = 1.0f;
    }
    return;
  }

  __shared__ float sA[DX * DX];
  __shared__ float sPt[DX * DX];
  __shared__ float sDP[DX * DX];
  __shared__ float sTmp[DX * DX];
  __shared__ float sPb[DX * DX];    // P_back carry
  __shared__ float sAug[DX * 33];   // [P_pred | A*P_t]
  __shared__ float sMuB[DX];
  __shared__ float sMuN[DX];
  __shared__ float sDv[DX];

  for (int i = lane; i < DX * DX; i += 32) sA[i] = g_A[i];
  {
    size_t mu_off = ((size_t)b * T_STEPS + (T_STEPS - 1)) * DX;
    size_t P_off  = ((size_t)b * T_STEPS + (T_STEPS - 1)) * DX * DX;
    if (lane < DX) {
      float v = o_mut[mu_off + lane];
      sMuB[lane] = v; o_muback[mu_off + lane] = v;
    }
#pragma unroll
    for (int r = 0; r < 8; ++r) {
      float v = o_Pt[P_off + lane * 8 + r];
      sPb[lane * 8 + r] = v; o_Pback[P_off + lane * 8 + r] = v;
    }
  }
  __syncthreads();

  const Frag4 aA = loadAfrag16(sA, DX, 0, 1.f, lane);  // invariant

  const v8f vzero = {};
#pragma unroll 1
  for (int t = T_STEPS - 2; t >= 0; --t) {
    size_t mu_off = ((size_t)b * T_STEPS + t) * DX;
    size_t P_off  = ((size_t)b * T_STEPS + t) * DX * DX;

#pragma unroll
    for (int r = 0; r < 8; ++r) {
      int idx = lane * 8 + r;
      sPt[idx] = o_Pt[P_off + idx];
      float pp = o_Ppred[P_off + idx];
      sDP[idx] = sPb[idx] - pp;                           // P_back - P_pred
      sAug[(idx >> 4) * 33 + (idx & 15)] = pp;            // aug cols 0..15
    }
    if (lane < DX) sDv[lane] = sMuB[lane] - o_mupred[mu_off + lane];
    __syncthreads();

    // AP = A * P_t -> aug cols 16..31
    const Frag4 bPt = loadBfrag16(sPt, DX, 0, lane);
    v8f ap = gemmFF(aA, bPt, vzero);
    stD16(&sAug[16], 33, ap, lane);
    __syncthreads();

    // Jordan: X = P_pred^{-1} (A P_t); J = X^T
#pragma unroll 1
    for (int p = 0; p < DX; ++p) {
      if (lane < DX) {
        float f = sAug[lane * 33 + p] / sAug[p * 33 + p];
        if (lane != p) {
#pragma unroll 4
          for (int c = p + 1; c < 32; ++c)
            sAug[lane * 33 + c] -= f * sAug[p * 33 + c];
        }
      }
      __syncthreads();
    }
    if (lane < DX) {
      float inv = 1.0f / sAug[lane * 33 + lane];
#pragma unroll
      for (int j = 0; j < DX; ++j) sAug[lane * 33 + 16 + j] *= inv;
    }
    __syncthreads();

    // mu_new = mu_t + J (mu_back - mu_pred)
    if (lane < DX) {
      float acc = 0.f;
#pragma unroll
      for (int k = 0; k < DX; ++k) acc += sAug[k * 33 + 16 + lane] * sDv[k];
      sMuN[lane] = o_mut[mu_off + lane] + acc;
    }

    // P_new = J * DP * J^T + P_t  (A-frag: J = X^T transposed read; B-frag: J^T = X)
    const Frag4 aJ  = loadAfrag16(&sAug[16], 33, 1, 1.f, lane);
    const Frag4 bDP = loadBfrag16(sDP, DX, 0, lane);
    v8f t2 = gemmFF(aJ, bDP, vzero);
    stD16(sTmp, DX, t2, lane);
    __syncthreads();
    const Frag4 aT2 = loadAfrag16(sTmp, DX, 0, 1.f, lane);
    const Frag4 bJT = loadBfrag16(&sAug[16], 33, 0, lane);
    v8f pnew = gemmFF(aT2, bJT, ldC16(sPt, DX, lane));
    __syncthreads();
    stD16(sPb, DX, pnew, lane);
    if (lane < DX) sMuB[lane] = sMuN[lane];
    __syncthreads();

    if (lane < DX) o_muback[mu_off + lane] = sMuB[lane];
#pragma unroll
    for (int r = 0; r < 8; ++r) o_Pback[P_off + lane * 8 + r] = sPb[lane * 8 + r];
    __syncthreads();
  }
}

extern "C" void kernel_launch(void* const* d_in, const int* in_sizes, int n_in,
                              void* d_out, int out_size, void* d_ws, size_t ws_size,
                              hipStream_t stream) {
  const float* a    = (const float*)d_in[0];
  const float* mask = (const float*)d_in[1];
  const float* A    = (const float*)d_in[2];
  const float* C    = (const float*)d_in[3];
  const float* mu0  = (const float*)d_in[4];
  const float* L0   = (const float*)d_in[5];
  const float* Wl   = (const float*)d_in[6];
  const float* Rl   = (const float*)d_in[7];
  const int*   flag = (const int*)d_in[8];

  float* out = (float*)d_out;
  const size_t MU = (size_t)B_SEQ * T_STEPS * DX;        // 2,048,000
  const size_t PM = (size_t)B_SEQ * T_STEPS * DX * DX;   // 32,768,000
  float* o_mupred = out;
  float* o_mut    = out + MU;
  float* o_muback = out + 2 * MU;
  float* o_Ppred  = out + 3 * MU;
  float* o_Pt     = out + 3 * MU + PM;
  float* o_Pback  = out + 3 * MU + 2 * PM;

  ldm_filter<<<B_SEQ, 32, 0, stream>>>(a, mask, A, C, mu0, L0, Wl, Rl,
                                       o_mupred, o_mut, o_Ppred, o_Pt);
  ldm_smooth<<<B_SEQ, 32, 0, stream>>>(A, flag, o_mupred, o_mut, o_Ppred, o_Pt,
                                       o_muback, o_Pback);
  (void)in_sizes; (void)n_in; (void)out_size; (void)d_ws; (void)ws_size;
}